// Model_6803228197638
// MI455X (gfx1250) — compile-verified
//
#include <hip/hip_runtime.h>
#include <hip/hip_bf16.h>
#include <math.h>
#include <stdint.h>

typedef __attribute__((ext_vector_type(2))) float v2f;
typedef __attribute__((ext_vector_type(8))) float v8f;
typedef __attribute__((ext_vector_type(4))) unsigned int u32x4;
typedef __attribute__((ext_vector_type(8))) unsigned int u32x8;

#define LDF 528          // 33 * 16 feature slots per node
#define EPS 1e-5f

static __device__ __forceinline__ v8f wmma4(v2f a, v2f b, v8f c) {
    // V_WMMA_F32_16X16X4_F32 : D = A(16x4) * B(4x16) + C(16x16), fp32 throughout
    return __builtin_amdgcn_wmma_f32_16x16x4_f32(
        /*neg_a=*/false, a, /*neg_b=*/false, b,
        /*c_mod=*/(short)0, c, /*reuse_a=*/false, /*reuse_b=*/false);
}

// TDM: DMA a flat run of `nbytes/4` dwords from global `ga` into LDS offset `lds_off`.
// 1-D tile: tensor_dim0 = tile_dim0 = nelem, stride0 = nelem, data_size = 4B.
static __device__ __forceinline__ void tdm_load_1d(unsigned long long ga,
                                                   unsigned int lds_off,
                                                   unsigned int nelem) {
    u32x4 g0;
    g0[0] = 1u;                                            // count=1, user descriptor
    g0[1] = lds_off;                                       // lds_addr (bytes)
    g0[2] = (unsigned int)(ga & 0xffffffffu);              // global_addr[31:0]
    g0[3] = (unsigned int)((ga >> 32) & 0x01ffffffu)       // global_addr[56:32]
            | 0x80000000u;                                 // type=2 ("image")
    u32x8 g1;
    g1[0] = 2u << 16;                                      // data_size = 4 bytes
    g1[1] = (nelem & 0xffffu) << 16;                       // tensor_dim0[15:0]
    g1[2] = ((nelem >> 16) & 0xffffu) | (1u << 16);        // tensor_dim0[31:16], tensor_dim1=1
    g1[3] = (nelem & 0xffffu) << 16;                       // tile_dim0 (<=16384 fits 16b)
    g1[4] = 1u;                                            // tile_dim1=1, tile_dim2=0
    g1[5] = nelem;                                         // tensor_dim0_stride[31:0]
    g1[6] = 0u;
    g1[7] = 0u;
    asm volatile("tensor_load_to_lds %0, %1" :: "s"(g0), "s"(g1) : "memory");
}

// ---------------------------------------------------------------- degrees ---
__global__ void k_deg_init(float* degA, float* degB, int n) {
    int i = blockIdx.x * blockDim.x + threadIdx.x;
    if (i < n) { degA[i] = 1.0f; degB[i] = 1.0f; }   // +1 self loop
}
__global__ void k_deg_count(const int* __restrict__ src, const int* __restrict__ dst,
                            int E, float* degA, float* degB) {
    int e = blockIdx.x * blockDim.x + threadIdx.x;
    if (e < E) {
        atomicAdd(&degA[dst[e]], 1.0f);   // deg over targets (source_to_target gcn)
        atomicAdd(&degB[src[e]], 1.0f);   // deg over sources (reversed gcn)
    }
}
__global__ void k_deg_rsqrt(float* degA, float* degB, int n) {
    int i = blockIdx.x * blockDim.x + threadIdx.x;
    if (i < n) { degA[i] = rsqrtf(degA[i]); degB[i] = rsqrtf(degB[i]); }
}

// ------------------------------------------------------------------ embed ---
__global__ void k_embed(const float* __restrict__ x, const float* __restrict__ W,
                        const float* __restrict__ b, float* feat, int n) {
    int t = blockIdx.x * blockDim.x + threadIdx.x;
    int v = t >> 4, c = t & 15;
    if (v >= n) return;
    float acc = b[c];
    #pragma unroll
    for (int k = 0; k < 4; ++k) acc += x[(size_t)v * 4 + k] * W[k * 16 + c];
    feat[(size_t)v * LDF + c] = acc;
}

// ------------------------------------------------ per-layer BN/bias fold ----
// s[c]=g*rsqrt(v+eps); beta quirk: beta = fc-bias broadcast (b[c mod 32]).
// fused bias' = b_fc + t @ W ; also fold conv-BN (32ch) into s2/t2.
__global__ void k_prep_layer(int C,
                             const float* __restrict__ g, const float* __restrict__ m,
                             const float* __restrict__ vv,
                             const float* __restrict__ Wfc, const float* __restrict__ bfc,
                             const float* __restrict__ g2, const float* __restrict__ b2,
                             const float* __restrict__ m2, const float* __restrict__ v2c,
                             float* sA, float* biasP, float* s2, float* t2) {
    __shared__ float tsh[512];
    int c = threadIdx.x;
    if (c < C) {
        float s = g[c] * rsqrtf(vv[c] + EPS);
        sA[c] = s;
        tsh[c] = bfc[c & 31] - m[c] * s;
    }
    __syncthreads();
    if (c < 32) {
        float acc = bfc[c];
        for (int k = 0; k < C; ++k) acc += tsh[k] * Wfc[(size_t)k * 32 + c];
        biasP[c] = acc;
        float s = g2[c] * rsqrtf(v2c[c] + EPS);
        s2[c] = s;
        t2[c] = b2[c] - m2[c] * s;
    }
}

// ----------------------------------- FC GEMM: y2 = relu(BN2(relu(A@W+b))) ---
// A = BN-scaled feat[:, 0:C]; each wave: 16 rows x 32 cols (two 16x16 tiles).
// W (C x 32) is DMA'd into LDS by the Tensor Data Mover once per block.
__global__ void k_fc_gemm(const float* __restrict__ feat, int C,
                          const float* __restrict__ W, const float* __restrict__ sA,
                          const float* __restrict__ biasP,
                          const float* __restrict__ s2, const float* __restrict__ t2,
                          float* __restrict__ y2, int n) {
    __shared__ float Wl[512 * 32];     // up to 64 KB (WGP has 320 KB LDS)
    __shared__ float ssc[512];

    if (threadIdx.x < 32) {            // wave 0 issues the TDM DMA (EXEC ignored)
        tdm_load_1d((unsigned long long)(uintptr_t)W,
                    (unsigned int)(uintptr_t)(&Wl[0]),
                    (unsigned int)(C * 32));
        __builtin_amdgcn_s_wait_tensorcnt(0);
    }
    for (int i = threadIdx.x; i < C; i += 256) ssc[i] = sA[i];
    __syncthreads();

    int wave = threadIdx.x >> 5, lane = threadIdx.x & 31;
    int row0 = (blockIdx.x * 8 + wave) * 16;
    int lh = lane >> 4, lm = lane & 15;
    int arow = row0 + lm;
    if (arow >= n) arow = n - 1;       // clamp: garbage rows only feed unstored outputs
    const float* Ar = feat + (size_t)arow * LDF;

    v8f c0 = {}; v8f c1 = {};
    for (int k0 = 0; k0 < C; k0 += 4) {
        int ka = k0 + 2 * lh;                 // A lanes 0-15: K{k0,k0+1}; 16-31: K{k0+2,k0+3}
        float2 av = *(const float2*)(Ar + ka);     // global_load_b64, 8B aligned
        v2f a;
        a.x = av.x * ssc[ka];
        a.y = av.y * ssc[ka + 1];
        v2f b0, b1;                           // B same K striping, N = lane 0-15 (from LDS)
        b0.x = Wl[ka * 32 + lm];
        b0.y = Wl[(ka + 1) * 32 + lm];
        b1.x = Wl[ka * 32 + 16 + lm];
        b1.y = Wl[(ka + 1) * 32 + 16 + lm];
        c0 = wmma4(a, b0, c0);
        c1 = wmma4(a, b1, c1);
    }
    #pragma unroll
    for (int r = 0; r < 8; ++r) {
        int orow = row0 + r + 8 * lh;         // C layout: vgpr r -> rows r / r+8
        if (orow < n) {
            float v0 = fmaxf(c0[r] + biasP[lm], 0.0f);
            v0 = fmaxf(v0 * s2[lm] + t2[lm], 0.0f);
            y2[(size_t)orow * 32 + lm] = v0;
            float v1 = fmaxf(c1[r] + biasP[16 + lm], 0.0f);
            v1 = fmaxf(v1 * s2[16 + lm] + t2[16 + lm], 0.0f);
            y2[(size_t)orow * 32 + 16 + lm] = v1;
        }
    }
}

// ------------------------- conv GEMM: h = y2 @ [Win | Wout], pre-scaled -----
__global__ void k_conv_gemm(const float* __restrict__ y2,
                            const float* __restrict__ Win, const float* __restrict__ Wout,
                            const float* __restrict__ dinvA, const float* __restrict__ dinvB,
                            float* __restrict__ hs, float* __restrict__ acc, int n) {
    __shared__ float Wc[32 * 16];      // [Win | Wout] interleaved, 2 KB
    for (int i = threadIdx.x; i < 512; i += 256) {
        int k = i >> 4, col = i & 15;
        Wc[i] = (col < 8) ? Win[k * 8 + col] : Wout[k * 8 + (col - 8)];
    }
    __syncthreads();

    int wave = threadIdx.x >> 5, lane = threadIdx.x & 31;
    int row0 = (blockIdx.x * 8 + wave) * 16;
    int lh = lane >> 4, lm = lane & 15;
    int arow = row0 + lm;
    if (arow >= n) arow = n - 1;       // clamp (see fc_gemm)
    const float* Ar = y2 + (size_t)arow * 32;

    v8f c = {};
    #pragma unroll
    for (int k0 = 0; k0 < 32; k0 += 4) {
        int ka = k0 + 2 * lh;
        float2 av = *(const float2*)(Ar + ka);
        v2f a; a.x = av.x; a.y = av.y;
        v2f b;
        b.x = Wc[ka * 16 + lm];
        b.y = Wc[(ka + 1) * 16 + lm];
        c = wmma4(a, b, c);
    }
    #pragma unroll
    for (int r = 0; r < 8; ++r) {
        int orow = row0 + r + 8 * lh;
        if (orow < n) {
            float d = (lm < 8) ? dinvA[orow] : dinvB[orow];
            hs[(size_t)orow * 16 + lm] = c[r] * d;   // pre-scaled message
            acc[(size_t)orow * 16 + lm] = 0.0f;      // zero accumulator for scatter
        }
    }
}

// ----------------------------------------------------------------- scatter --
__global__ void k_scatter(const int* __restrict__ src, const int* __restrict__ dst,
                          int E, const float* __restrict__ hs, float* __restrict__ acc) {
    int t = blockIdx.x * blockDim.x + threadIdx.x;
    int e = t >> 1, dir = t & 1;
    if (e >= E) return;
    if (e + 4096 < E) {                       // stream edge indices through L1/L2
        __builtin_prefetch(&src[e + 4096], 0, 1);
        __builtin_prefetch(&dst[e + 4096], 0, 1);
    }
    int u, v, base;
    if (dir == 0) { u = src[e]; v = dst[e]; base = 0; }   // source_to_target half
    else          { u = dst[e]; v = src[e]; base = 8; }   // reversed half
    const float4* hp = (const float4*)(hs + (size_t)u * 16 + base);
    float4 h0 = hp[0], h1 = hp[1];
    float* ap = acc + (size_t)v * 16 + base;
    atomicAdd(ap + 0, h0.x); atomicAdd(ap + 1, h0.y);
    atomicAdd(ap + 2, h0.z); atomicAdd(ap + 3, h0.w);
    atomicAdd(ap + 4, h1.x); atomicAdd(ap + 5, h1.y);
    atomicAdd(ap + 6, h1.z); atomicAdd(ap + 7, h1.w);
}

// --------------------------------------------------------------- finalize ---
// out = dinv * (acc + hs) + bias  (self loop folds in since hs is pre-scaled)
__global__ void k_finalize(const float* __restrict__ acc, const float* __restrict__ hs,
                           const float* __restrict__ dinvA, const float* __restrict__ dinvB,
                           const float* __restrict__ bin, const float* __restrict__ bout,
                           float* feat, int slot, int n) {
    int t = blockIdx.x * blockDim.x + threadIdx.x;
    int v = t >> 4, c = t & 15;
    if (v >= n) return;
    float d = (c < 8) ? dinvA[v] : dinvB[v];
    float b = (c < 8) ? bin[c] : bout[c - 8];
    feat[(size_t)v * LDF + slot + c] =
        d * (acc[(size_t)v * 16 + c] + hs[(size_t)v * 16 + c]) + b;
}

// ------------------------------------------------------------------- pool ---
__global__ void k_graph_starts(const int* __restrict__ batch, int n, int* starts) {
    int g = threadIdx.x;
    if (g > 64) return;
    int lo = 0, hi = n;
    while (lo < hi) { int mid = (lo + hi) >> 1; if (batch[mid] < g) lo = mid + 1; else hi = mid; }
    starts[g] = lo;
}
__global__ void k_pool(const float* __restrict__ feat, const int* __restrict__ starts,
                       const float* __restrict__ g, const float* __restrict__ b,
                       const float* __restrict__ m, const float* __restrict__ v,
                       float* __restrict__ gout) {
    int gi = blockIdx.x;
    int s = starts[gi], e = starts[gi + 1];
    for (int ci = threadIdx.x; ci < 512; ci += blockDim.x) {
        float sc = g[ci] * rsqrtf(v[ci] + EPS);
        float sh = b[ci] - m[ci] * sc;
        float mx = -3.402823466e38f, sum = 0.0f;
        for (int i = s; i < e; ++i) {
            float val = feat[(size_t)i * LDF + 16 + ci] * sc + sh;
            mx = fmaxf(mx, val);
            sum += val;
        }
        float mean = (e > s) ? sum / (float)(e - s) : 0.0f;
        if (e <= s) mx = 0.0f;                       // isfinite guard for empty graph
        gout[(size_t)gi * 1024 + ci] = mx;
        gout[(size_t)gi * 1024 + 512 + ci] = mean;
    }
}

// ----------------------------------------------------------- output head ----
__global__ void k_bn_scale(const float* __restrict__ g, const float* __restrict__ beta,
                           const float* __restrict__ m, const float* __restrict__ v,
                           int C, float* s, float* t) {
    int i = blockIdx.x * blockDim.x + threadIdx.x;
    if (i < C) { float sc = g[i] * rsqrtf(v[i] + EPS); s[i] = sc; t[i] = beta[i] - m[i] * sc; }
}
__global__ void k_fuse_bias(const float* __restrict__ t, const float* __restrict__ W,
                            const float* __restrict__ bias, int C, int N, float* biasP) {
    int j = blockIdx.x * blockDim.x + threadIdx.x;
    if (j >= N) return;
    float acc = bias[j];
    for (int k = 0; k < C; ++k) acc += t[k] * W[(size_t)k * N + j];
    biasP[j] = acc;
}
// M=64 rows, generic C,N multiples of 16; relu output.
__global__ void k_out_gemm(const float* __restrict__ A, int C,
                           const float* __restrict__ W, int N,
                           const float* __restrict__ sA, const float* __restrict__ biasP,
                           float* __restrict__ Y) {
    int wave = threadIdx.x >> 5, lane = threadIdx.x & 31;
    int tile = blockIdx.x * 8 + wave;
    int ntiles = N >> 4;
    int row0 = (tile / ntiles) * 16, col0 = (tile % ntiles) * 16;
    int lh = lane >> 4, lm = lane & 15;
    const float* Ar = A + (size_t)(row0 + lm) * C;
    v8f c = {};
    for (int k0 = 0; k0 < C; k0 += 4) {
        int ka = k0 + 2 * lh;
        float2 av = *(const float2*)(Ar + ka);
        v2f a; a.x = av.x * sA[ka]; a.y = av.y * sA[ka + 1];
        v2f b; b.x = W[(size_t)ka * N + col0 + lm];
               b.y = W[(size_t)(ka + 1) * N + col0 + lm];
        c = wmma4(a, b, c);
    }
    #pragma unroll
    for (int r = 0; r < 8; ++r) {
        int orow = row0 + r + 8 * lh;
        Y[(size_t)orow * N + col0 + lm] = fmaxf(c[r] + biasP[col0 + lm], 0.0f);
    }
}
__global__ void k_out_lin(const float* __restrict__ A, const float* __restrict__ W,
                          const float* __restrict__ b, float* out) {
    __shared__ float red[256];
    int gi = blockIdx.x;
    float acc = 0.0f;
    for (int k = threadIdx.x; k < 1024; k += 256) acc += A[(size_t)gi * 1024 + k] * W[k];
    red[threadIdx.x] = acc;
    __syncthreads();
    for (int s = 128; s > 0; s >>= 1) {
        if (threadIdx.x < s) red[threadIdx.x] += red[threadIdx.x + s];
        __syncthreads();
    }
    if (threadIdx.x == 0) out[gi] = red[0] + b[0];
}

// ==================================================================== host ==
extern "C" void kernel_launch(void* const* d_in, const int* in_sizes, int n_in,
                              void* d_out, int out_size, void* d_ws, size_t ws_size,
                              hipStream_t stream) {
    const int n = in_sizes[0] / 4;          // N_NODES
    const int E = in_sizes[1] / 2;          // N_EDGES
    const float* x     = (const float*)d_in[0];
    const int*   src   = (const int*)d_in[1];
    const int*   dst   = src + E;
    const int*   batch = (const int*)d_in[2];

    // params flattened via sorted-key pytree order:
    // conv[l]: Win,Wout,b,bin,bout,g,m,v  -> base 3 + 8l
    // embed: W(259), b(260)
    // fc[l]: W,b,g,m,v -> base 261 + 5l
    // out_fc1: 421..425 (W,b,g,m,v) ; out_fc2: 426..430
    // out_lin: W(431), b(432) ; pool_bn: b(433), g(434), m(435), v(436)
    auto F = [&](int i) { return (const float*)d_in[i]; };

    float* wsf = (float*)d_ws;
    size_t o = 0;
    float* feat  = wsf + o; o += (size_t)n * LDF;
    float* y2    = wsf + o; o += (size_t)n * 32;
    float* hs    = wsf + o; o += (size_t)n * 16;
    float* acc   = wsf + o; o += (size_t)n * 16;
    float* dinvA = wsf + o; o += n;
    float* dinvB = wsf + o; o += n;
    float* sA    = wsf + o; o += 512;
    float* biasP = wsf + o; o += 32;
    float* s2    = wsf + o; o += 32;
    float* t2    = wsf + o; o += 32;
    float* gpool = wsf + o; o += 64 * 1024;
    float* g1    = wsf + o; o += 64 * 1024;
    float* g2    = wsf + o; o += 64 * 1024;
    float* sO    = wsf + o; o += 1024;
    float* tO    = wsf + o; o += 1024;
    float* bO    = wsf + o; o += 1024;
    int*   starts = (int*)(wsf + o); o += 65;

    const int nbN   = (n + 255) / 256;
    const int nbE   = (E + 255) / 256;
    const int nbRow = (n + 127) / 128;       // 8 waves/block, 16 rows/wave

    // degrees (edge-structure is layer-invariant: compute once)
    k_deg_init<<<nbN, 256, 0, stream>>>(dinvA, dinvB, n);
    k_deg_count<<<nbE, 256, 0, stream>>>(src, dst, E, dinvA, dinvB);
    k_deg_rsqrt<<<nbN, 256, 0, stream>>>(dinvA, dinvB, n);

    // input embed -> feat[:, 0:16]
    k_embed<<<(n * 16 + 255) / 256, 256, 0, stream>>>(x, F(259), F(260), feat, n);

    for (int l = 0; l < 32; ++l) {
        const int C = 16 * (l + 1);
        const int fcb = 261 + 5 * l, cvb = 3 + 8 * l;
        const float* fcW = F(fcb), *fcBias = F(fcb + 1), *fcG = F(fcb + 2),
                   * fcM = F(fcb + 3), *fcV = F(fcb + 4);
        const float* Win = F(cvb), *Wout = F(cvb + 1), *cvB = F(cvb + 2),
                   * bin = F(cvb + 3), *bout = F(cvb + 4), *cvG = F(cvb + 5),
                   * cvM = F(cvb + 6), *cvV = F(cvb + 7);

        k_prep_layer<<<1, 512, 0, stream>>>(C, fcG, fcM, fcV, fcW, fcBias,
                                            cvG, cvB, cvM, cvV, sA, biasP, s2, t2);
        k_fc_gemm<<<nbRow, 256, 0, stream>>>(feat, C, fcW, sA, biasP, s2, t2, y2, n);
        k_conv_gemm<<<nbRow, 256, 0, stream>>>(y2, Win, Wout, dinvA, dinvB, hs, acc, n);
        k_scatter<<<(2 * E + 255) / 256, 256, 0, stream>>>(src, dst, E, hs, acc);
        k_finalize<<<(n * 16 + 255) / 256, 256, 0, stream>>>(acc, hs, dinvA, dinvB,
                                                             bin, bout, feat, C, n);
    }

    // pooling (BN applied per element before max/mean)
    k_graph_starts<<<1, 128, 0, stream>>>(batch, n, starts);
    k_pool<<<64, 256, 0, stream>>>(feat, starts, F(434), F(433), F(435), F(436), gpool);

    // head: fc1
    k_bn_scale<<<4, 256, 0, stream>>>(F(423), F(422), F(424), F(425), 1024, sO, tO);
    k_fuse_bias<<<4, 256, 0, stream>>>(tO, F(421), F(422), 1024, 1024, bO);
    k_out_gemm<<<32, 256, 0, stream>>>(gpool, 1024, F(421), 1024, sO, bO, g1);
    // head: fc2
    k_bn_scale<<<4, 256, 0, stream>>>(F(428), F(427), F(429), F(430), 1024, sO, tO);
    k_fuse_bias<<<4, 256, 0, stream>>>(tO, F(426), F(427), 1024, 1024, bO);
    k_out_gemm<<<32, 256, 0, stream>>>(g1, 1024, F(426), 1024, sO, bO, g2);
    // final linear -> d_out[64]
    k_out_lin<<<64, 256, 0, stream>>>(g2, F(431), F(432), (float*)d_out);

    (void)n_in; (void)out_size; (void)ws_size;
}